// SAGE_5789615915673
// MI455X (gfx1250) — compile-verified
//
#include <hip/hip_runtime.h>

// ---------------------------------------------------------------------------
// GraphSAGE 2-layer forward for MI455X (gfx1250, wave32).
//   L0: h = relu(x[:ND0] @ Ws0 + mean0 @ Wn0 + b0)      (40960 x 256)
//   L1: out = h[:ND1] @ Ws1 + mean1 @ Wn1 + b1          (4096 x 47)
// GEMMs use V_WMMA_F32_16X16X4_F32 (exact fp32, matches reference).
// Edge aggregation uses hardware global_atomic_add_f32 into L2-resident sums.
// Workload is memory-bound (~0.5-0.9 GB HBM traffic => ~25-40us @ 23.3 TB/s);
// GEMM is 10.9 GFLOP fp32, far under the WMMA ceiling, so exact f32 WMMA is
// the right precision (no conversion cost, bit-matches reference).
// ---------------------------------------------------------------------------

typedef float v2f __attribute__((ext_vector_type(2)));
typedef float v8f __attribute__((ext_vector_type(8)));

#define NS0 409600
#define ND0 40960
#define NS1 40960
#define ND1 4096
#define NE0 409600
#define NE1 40960
#define DIN 256
#define DH  256
#define DOUT 47

// ---------------------------- zero workspace -------------------------------
__global__ void sage_zero_kernel(float4* __restrict__ p, int n4) {
    int i = blockIdx.x * blockDim.x + threadIdx.x;
    if (i < n4) p[i] = make_float4(0.f, 0.f, 0.f, 0.f);
}

// ------------------------ edge gather/scatter-add --------------------------
// 64 threads per edge; each handles 4 contiguous feature floats.
__global__ void sage_scatter_kernel(const float* __restrict__ feat,
                                    const int* __restrict__ src,
                                    const int* __restrict__ dst,
                                    float* __restrict__ sums,
                                    float* __restrict__ cnt,
                                    int E) {
    long long tid = (long long)blockIdx.x * blockDim.x + threadIdx.x;
    if (tid >= (long long)E * 64) return;
    int e = (int)(tid >> 6);
    int c = ((int)tid & 63) << 2;           // feature offset 0..252
    int s = src[e];
    int d = dst[e];
    float4 v = *(const float4*)(feat + (size_t)s * 256 + c);
    float* p = sums + (size_t)d * 256 + c;
    unsafeAtomicAdd(p + 0, v.x);
    unsafeAtomicAdd(p + 1, v.y);
    unsafeAtomicAdd(p + 2, v.z);
    unsafeAtomicAdd(p + 3, v.w);
    if (c == 0) unsafeAtomicAdd(cnt + d, 1.0f);
}

// ----------------------------- WMMA GEMM -----------------------------------
__device__ __forceinline__ v8f wmma_f32(v2f a, v2f b, v8f c) {
    // D = A(16x4,f32) * B(4x16,f32) + C(16x16,f32)
    return __builtin_amdgcn_wmma_f32_16x16x4_f32(
        /*neg_a=*/false, a, /*neg_b=*/false, b,
        /*c_mod=*/(short)0, c, /*reuse_a=*/false, /*reuse_b=*/false);
}

// One wave per 16x16 output tile.
//   out[m, n] = Aself[m,:] @ Bs[:,n] + (Sneigh[m,:] / max(cnt[m],1)) @ Bn[:,n] + bias[n]
// A row stride fixed at 256, K fixed at 256. B row stride == NREAL.
// Padded lanes (col >= NREAL) clamp their column: they compute a duplicate of
// column NREAL-1 (WMMA columns are independent) and simply never store it.
// This keeps EXEC all-ones with zero divergent branches in the hot loop.
template <int NREAL, int NTILES, bool RELU>
__global__ void sage_wmma_gemm_kernel(const float* __restrict__ Aself,
                                      const float* __restrict__ Sneigh,
                                      const float* __restrict__ cnt,
                                      const float* __restrict__ Bs,
                                      const float* __restrict__ Bn,
                                      const float* __restrict__ bias,
                                      float* __restrict__ out) {
    constexpr int K = 256;
    constexpr int LDA = 256;
    int wave = blockIdx.x * (blockDim.x >> 5) + (threadIdx.x >> 5);
    int tm = wave / NTILES;
    int tn = wave - tm * NTILES;
    int lane = threadIdx.x & 31;
    int half = lane >> 4;                   // 0: lanes 0-15, 1: lanes 16-31
    int lidx = lane & 15;
    int m0 = tm << 4;
    int arow = m0 + lidx;                   // A: row m = lane&15 (both halves)
    int col  = (tn << 4) + lidx;            // B/C: col n = lane&15
    int colc = (col < NREAL) ? col : (NREAL - 1);   // clamped (always in-bounds)
    int koff = half << 1;                   // A/B k offset: 0 or 2

    const float* aS = Aself  + (size_t)arow * LDA + koff;
    const float* aN = Sneigh + (size_t)arow * LDA + koff;
    const float* bS = Bs + (size_t)koff * NREAL + colc;
    const float* bN = Bn + (size_t)koff * NREAL + colc;
    float invc = 1.0f / fmaxf(cnt[arow], 1.0f);   // fold mean into A fragment

    v8f acc = {0.f, 0.f, 0.f, 0.f, 0.f, 0.f, 0.f, 0.f};

#pragma unroll 16
    for (int k = 0; k < K; k += 4) {
        v2f a_s = *(const v2f*)(aS + k);          // A[m, k+koff .. k+koff+1]
        v2f a_n = *(const v2f*)(aN + k);
        a_n.x *= invc;
        a_n.y *= invc;
        v2f b_s, b_n;                              // immediate-offset loads
        b_s.x = bS[(size_t)k * NREAL];
        b_s.y = bS[(size_t)(k + 1) * NREAL];
        b_n.x = bN[(size_t)k * NREAL];
        b_n.y = bN[(size_t)(k + 1) * NREAL];
        acc = wmma_f32(a_s, b_s, acc);
        acc = wmma_f32(a_n, b_n, acc);
    }

    float bv = bias[colc];
#pragma unroll
    for (int r = 0; r < 8; ++r) {
        float v = acc[r] + bv;
        if (RELU) v = fmaxf(v, 0.f);
        int m = m0 + (half << 3) + r;             // C: m = 8*half + r
        if (col < NREAL) out[(size_t)m * NREAL + col] = v;
    }
}

// ------------------------------- launcher ----------------------------------
extern "C" void kernel_launch(void* const* d_in, const int* in_sizes, int n_in,
                              void* d_out, int out_size, void* d_ws, size_t ws_size,
                              hipStream_t stream) {
    const float* x    = (const float*)d_in[0];
    const int*   src0 = (const int*)  d_in[1];
    const int*   dst0 = (const int*)  d_in[2];
    const int*   src1 = (const int*)  d_in[3];
    const int*   dst1 = (const int*)  d_in[4];
    const float* Ws0  = (const float*)d_in[5];
    const float* Wn0  = (const float*)d_in[6];
    const float* b0   = (const float*)d_in[7];
    const float* Ws1  = (const float*)d_in[8];
    const float* Wn1  = (const float*)d_in[9];
    const float* b1   = (const float*)d_in[10];
    float* out = (float*)d_out;

    // Workspace layout (floats): sums0 | cnt0 | sums1 | cnt1 | h
    float* ws    = (float*)d_ws;
    float* sums0 = ws;
    float* cnt0  = sums0 + (size_t)ND0 * DH;
    float* sums1 = cnt0 + ND0;
    float* cnt1  = sums1 + (size_t)ND1 * DH;
    float* h     = cnt1 + ND1;

    // 1) zero sums/cnt region (contiguous, multiple of 4 floats)
    long long nzero = (long long)ND0 * DH + ND0 + (long long)ND1 * DH + ND1;
    int n4 = (int)(nzero >> 2);
    sage_zero_kernel<<<(n4 + 255) / 256, 256, 0, stream>>>((float4*)ws, n4);

    // 2) layer-0 edge aggregation: sums0/cnt0 += scatter(x[src0] -> dst0)
    {
        long long total = (long long)NE0 * 64;
        int blocks = (int)((total + 255) / 256);
        sage_scatter_kernel<<<blocks, 256, 0, stream>>>(x, src0, dst0, sums0, cnt0, NE0);
    }

    // 3) layer-0 fused GEMM + bias + relu -> h  (40960 x 256)
    {
        int waves = (ND0 / 16) * (DH / 16);   // 2560 * 16 = 40960 tiles
        int blocks = waves / 8;               // 8 waves (256 threads) per block
        sage_wmma_gemm_kernel<DH, DH / 16, true><<<blocks, 256, 0, stream>>>(
            x, sums0, cnt0, Ws0, Wn0, b0, h);
    }

    // 4) layer-1 edge aggregation on h
    {
        long long total = (long long)NE1 * 64;
        int blocks = (int)((total + 255) / 256);
        sage_scatter_kernel<<<blocks, 256, 0, stream>>>(h, src1, dst1, sums1, cnt1, NE1);
    }

    // 5) layer-1 GEMM + bias -> out (4096 x 47, N padded to 3 tiles of 16)
    {
        constexpr int ntiles = (DOUT + 15) / 16;  // 3
        int waves = (ND1 / 16) * ntiles;          // 768
        int blocks = waves / 8;                   // 96
        sage_wmma_gemm_kernel<DOUT, ntiles, false><<<blocks, 256, 0, stream>>>(
            h, sums1, cnt1, Ws1, Wn1, b1, out);
    }
}